// LIF_69277822485149
// MI455X (gfx1250) — compile-verified
//
#include <hip/hip_runtime.h>

#ifndef __has_builtin
#define __has_builtin(x) 0
#endif

#if __has_builtin(__builtin_amdgcn_global_load_async_to_lds_b64)
#define LIF_ASYNC 1
#else
#define LIF_ASYNC 0
#endif

#define BLOCK   256
#define STAGES  16                // async pipeline depth (power of two)
#define VEC     2                 // neurons per thread
#define NEUR    (32 * 64 * 64)    // B*H*W, fixed by harness
#define TSTEPS  256               // T, fixed by harness
#define STRIDE2 (NEUR / VEC)      // v2f elements between time steps (65536)

// Native vector types (HIP's float2 is a class; builtins need real vectors).
typedef float v2f __attribute__((ext_vector_type(2)));
typedef int   v2i __attribute__((vector_size(8)));
typedef __attribute__((address_space(1))) v2i* gptr_v2i;
typedef __attribute__((address_space(3))) v2i* lptr_v2i;

// One LIF update for a single neuron. Matches reference op order:
//   v_dec = v + dt*tau_mem_inv*((0 - v) + i)
//   i_dec = i * (1 - dt*tau_syn_inv)
//   z     = (v_dec - 1 > 0) ? 1 : 0
//   v'    = (1-z)*v_dec ; i' = i_dec + x
__device__ __forceinline__ float lif1(float xv, float& v, float& i,
                                      float alpha, float beta) {
  float vd = v + alpha * ((0.0f - v) + i);
  float id = i * beta;
  bool  s  = (vd - 1.0f) > 0.0f;
  float z  = s ? 1.0f : 0.0f;
  v = s ? 0.0f : vd;
  i = id + xv;
  return z;
}

__global__ __launch_bounds__(BLOCK) void LIF_69277822485149_kernel(
    const float* __restrict__ x,
    const float* __restrict__ tau_syn_inv,
    const float* __restrict__ tau_mem_inv,
    float* __restrict__ out)
{
  const float alpha = 0.001f * tau_mem_inv[0];           // dt * tau_mem_inv
  const float beta  = 1.0f - 0.001f * tau_syn_inv[0];    // 1 - dt * tau_syn_inv

  const int tid   = threadIdx.x;
  const int pair0 = blockIdx.x * BLOCK + tid;            // v2f index of this thread
  const v2f* __restrict__ xg = (const v2f*)x + pair0;
  v2f* __restrict__ og = (v2f*)out + pair0;

  float vx = 0.0f, vy = 0.0f, ix = 0.0f, iy = 0.0f;

#if LIF_ASYNC
  // -------- async global->LDS streaming pipeline (ASYNCcnt-tracked) --------
  __shared__ v2f buf[STAGES][BLOCK];   // 32 KB of 320 KB/WGP

  // Prologue: fill all pipeline stages (stage offsets are compile-time).
  #pragma unroll
  for (int s = 0; s < STAGES; ++s) {
    __builtin_amdgcn_global_load_async_to_lds_b64(
        (gptr_v2i)(xg + s * STRIDE2), (lptr_v2i)(&buf[s][tid]), 0, 0);
  }

  // Steady state: exactly STAGES loads in flight at all times; loads
  // complete in order, so asynccnt <= STAGES-1 means "oldest landed".
  for (int t = 0; t < TSTEPS - STAGES; ++t) {
    asm volatile("s_wait_asynccnt %0" :: "n"(STAGES - 1) : "memory");
    const int st = t & (STAGES - 1);
    v2f xv = buf[st][tid];                    // ds_load_b64, own lane's slot

    v2f z;
    z.x = lif1(xv.x, vx, ix, alpha, beta);
    z.y = lif1(xv.y, vy, iy, alpha, beta);
    __builtin_nontemporal_store(z, og + t * STRIDE2);

    // Our LDS read has retired (value consumed); safe to overwrite the slot.
    asm volatile("s_wait_dscnt 0" ::: "memory");
    __builtin_amdgcn_global_load_async_to_lds_b64(
        (gptr_v2i)(xg + (t + STAGES) * STRIDE2), (lptr_v2i)(&buf[st][tid]), 0, 0);
  }

  // Drain: all remaining stages are already in flight; one wait retires
  // them all, then the last STAGES steps are LDS-only.
  asm volatile("s_wait_asynccnt 0" ::: "memory");
  for (int t = TSTEPS - STAGES; t < TSTEPS; ++t) {
    v2f xv = buf[t & (STAGES - 1)][tid];

    v2f z;
    z.x = lif1(xv.x, vx, ix, alpha, beta);
    z.y = lif1(xv.y, vy, iy, alpha, beta);
    __builtin_nontemporal_store(z, og + t * STRIDE2);
  }
#else
  // -------- fallback: register prefetch pipeline (LOADcnt-tracked) --------
  v2f xb[8];
  #pragma unroll
  for (int s = 0; s < 8; ++s)
    xb[s] = __builtin_nontemporal_load(xg + s * STRIDE2);
  for (int t = 0; t < TSTEPS; ++t) {
    const int st = t & 7;
    v2f xv = xb[st];
    int tn = t + 8;
    if (tn < TSTEPS)
      xb[st] = __builtin_nontemporal_load(xg + tn * STRIDE2);

    v2f z;
    z.x = lif1(xv.x, vx, ix, alpha, beta);
    z.y = lif1(xv.y, vy, iy, alpha, beta);
    __builtin_nontemporal_store(z, og + t * STRIDE2);
  }
#endif
}

extern "C" void kernel_launch(void* const* d_in, const int* in_sizes, int n_in,
                              void* d_out, int out_size, void* d_ws, size_t ws_size,
                              hipStream_t stream) {
  (void)n_in; (void)d_ws; (void)ws_size; (void)out_size; (void)in_sizes;
  const float* x   = (const float*)d_in[0];
  const float* tsi = (const float*)d_in[1];
  const float* tmi = (const float*)d_in[2];
  float* out = (float*)d_out;

  // Shapes fixed by the harness: x is [256, 32, 64, 64] fp32.
  const int threads = NEUR / VEC;                    // 65536
  dim3 grid(threads / BLOCK);                        // 256 blocks x 8 waves
  LIF_69277822485149_kernel<<<grid, BLOCK, 0, stream>>>(x, tsi, tmi, out);
}